// MultilayerAttentionIterativeDecoder_77051713290408
// MI455X (gfx1250) — compile-verified
//
#include <hip/hip_runtime.h>
#include <math.h>

// ---------------- problem constants ----------------
#define Dh   512
#define Vv   4000
#define Bb   64
#define Tt   256
#define Ss   512
#define D2   (2*Dh)    // 1024
#define D3   (3*Dh)    // 1536

typedef __attribute__((ext_vector_type(2))) float v2f;
typedef __attribute__((ext_vector_type(8))) float v8f;

// ---------------------------------------------------------------------------
// V_WMMA_F32_16X16X4_F32 fragment mapping (ISA 7.12.2):
//   A 16x4 : lane l -> A[M = l%16, K = 2*(l/16) + {0,1}]  (one float2 load)
//   B 4x16 : lane l -> W[N = l%16, K = 2*(l/16) + {0,1}]  (one float2 load)
//   C/D    : lane l, vgpr v -> C[M = v + 8*(l/16), N = l%16]
// ---------------------------------------------------------------------------
__device__ __forceinline__ v8f wmma4(v2f a, v2f b, v8f c) {
  return __builtin_amdgcn_wmma_f32_16x16x4_f32(false, a, false, b,
                                               (short)0, c, false, false);
}

// single 16x16 tile: C += A[16xK] * W[16xK]^T
__device__ __forceinline__ void wmma_acc_1x1(v8f& acc,
                                             const float* __restrict__ A, int lda,
                                             const float* __restrict__ W, int ldw,
                                             int K) {
  unsigned lane = threadIdx.x & 31u;
  unsigned half = lane >> 4;
  unsigned id16 = lane & 15u;
  const float* a = A + (size_t)id16 * lda + 2u * half;
  const float* w = W + (size_t)id16 * ldw + 2u * half;
  for (int k = 0; k < K; k += 4) {
    v2f av = *(const v2f*)(a + k);
    v2f bv = *(const v2f*)(w + k);
    acc = wmma4(av, bv, acc);
  }
}

// 16 x 64 tile: one A fragment reused across 4 adjacent N-subtiles
__device__ __forceinline__ void wmma_acc_1x4(v8f acc[4],
                                             const float* __restrict__ A, int lda,
                                             const float* __restrict__ W, int ldw,
                                             int K) {
  unsigned lane = threadIdx.x & 31u;
  unsigned half = lane >> 4;
  unsigned id16 = lane & 15u;
  const float* a = A + (size_t)id16 * lda + 2u * half;
  const float* w = W + (size_t)id16 * ldw + 2u * half;
  const size_t wstep = (size_t)16 * ldw;
  for (int k = 0; k < K; k += 4) {
    v2f av = *(const v2f*)(a + k);
    v2f b0 = *(const v2f*)(w + k);
    v2f b1 = *(const v2f*)(w + wstep + k);
    v2f b2 = *(const v2f*)(w + 2 * wstep + k);
    v2f b3 = *(const v2f*)(w + 3 * wstep + k);
    acc[0] = wmma4(av, b0, acc[0]);
    acc[1] = wmma4(av, b1, acc[1]);
    acc[2] = wmma4(av, b2, acc[2]);
    acc[3] = wmma4(av, b3, acc[3]);
  }
}

// 64 x 32 tile: 4 M-subtiles x 2 N-subtiles (A reused 2x, B reused 4x)
__device__ __forceinline__ void wmma_acc_4x2(v8f acc[4][2],
                                             const float* __restrict__ A, int lda,
                                             const float* __restrict__ W, int ldw,
                                             int K) {
  unsigned lane = threadIdx.x & 31u;
  unsigned half = lane >> 4;
  unsigned id16 = lane & 15u;
  const float* a = A + (size_t)id16 * lda + 2u * half;
  const float* w = W + (size_t)id16 * ldw + 2u * half;
  const size_t astep = (size_t)16 * lda;
  const size_t wstep = (size_t)16 * ldw;
  for (int k = 0; k < K; k += 4) {
    v2f a0 = *(const v2f*)(a + k);
    v2f a1 = *(const v2f*)(a + astep + k);
    v2f a2 = *(const v2f*)(a + 2 * astep + k);
    v2f a3 = *(const v2f*)(a + 3 * astep + k);
    v2f b0 = *(const v2f*)(w + k);
    v2f b1 = *(const v2f*)(w + wstep + k);
    acc[0][0] = wmma4(a0, b0, acc[0][0]);
    acc[1][0] = wmma4(a1, b0, acc[1][0]);
    acc[2][0] = wmma4(a2, b0, acc[2][0]);
    acc[3][0] = wmma4(a3, b0, acc[3][0]);
    acc[0][1] = wmma4(a0, b1, acc[0][1]);
    acc[1][1] = wmma4(a1, b1, acc[1][1]);
    acc[2][1] = wmma4(a2, b1, acc[2][1]);
    acc[3][1] = wmma4(a3, b1, acc[3][1]);
  }
}

__device__ __forceinline__ void wmma_store_bias(const v8f& acc,
                                                float* __restrict__ C, int ldc,
                                                const float* __restrict__ bias /* +col0 */) {
  unsigned lane = threadIdx.x & 31u;
  unsigned half = lane >> 4;
  unsigned id16 = lane & 15u;
  float bv = bias[id16];
#pragma unroll
  for (int v = 0; v < 8; ++v) {
    C[(size_t)(v + 8u * half) * ldc + id16] = acc[v] + bv;
  }
}

// ---------------------------------------------------------------------------
// Kernel 0: zero carry state (h, ctx)
// ---------------------------------------------------------------------------
__global__ void init_zero_kernel(float* __restrict__ p, int n) {
  int i = blockIdx.x * blockDim.x + threadIdx.x;
  if (i < n) p[i] = 0.0f;
}

// ---------------------------------------------------------------------------
// Kernel 1: keys/values = features @ {Wk,Wv}^T + {bk,bv}
//   [M = B*S = 32768, K = D] ; wave computes 16x64 (1x4 N-group, A reused 4x)
// ---------------------------------------------------------------------------
__global__ void kv_proj_kernel(const float* __restrict__ feat,
                               const float* __restrict__ Wk, const float* __restrict__ bk,
                               const float* __restrict__ Wv, const float* __restrict__ bv,
                               float* __restrict__ keys, float* __restrict__ vals) {
  const int MT = (Bb * Ss) / 16;  // 2048
  const int NG = Dh / 64;         // 8 groups of 4 N-tiles
  int gw = blockIdx.x * (blockDim.x >> 5) + (threadIdx.x >> 5);
  int matsel = gw / (MT * NG);
  int rem    = gw % (MT * NG);
  int mt = rem / NG, ng = rem % NG;

  const float* W  = matsel ? Wv : Wk;
  const float* bi = matsel ? bv : bk;
  float*       C  = matsel ? vals : keys;

  v8f acc[4] = {{}, {}, {}, {}};
  wmma_acc_1x4(acc, feat + (size_t)(mt * 16) * Dh, Dh,
               W + (size_t)(ng * 64) * Dh, Dh, Dh);
#pragma unroll
  for (int j = 0; j < 4; ++j) {
    int col0 = ng * 64 + j * 16;
    wmma_store_bias(acc[j], C + (size_t)(mt * 16) * Dh + col0, Dh, bi + col0);
  }
}

// ---------------------------------------------------------------------------
// Kernel 2: embedding gather  emb[b,t,:] = emb_W[input[b,t], :]
// ---------------------------------------------------------------------------
__global__ void gather_emb_kernel(const int* __restrict__ input,
                                  const float* __restrict__ embW,
                                  float* __restrict__ emb) {
  int i = blockIdx.x * blockDim.x + threadIdx.x;   // over B*T*D
  int row = i / Dh;                                 // b*T + t
  int d   = i - row * Dh;
  int tok = input[row];
  emb[i] = embW[(size_t)tok * Dh + d];
}

// ---------------------------------------------------------------------------
// Kernel 3 (per step): gi = [emb_t, ctx] @ W_ih^T + b_ih ; gh = h @ W_hh^T + b_hh
//   768 single tiles, 1 wave/tile (max parallelism along the sequential path)
// ---------------------------------------------------------------------------
__global__ void gate_gemm_kernel(const float* __restrict__ emb,   // [B,T,D]
                                 const float* __restrict__ ctx,   // [B,D]
                                 const float* __restrict__ h,     // [B,D]
                                 const float* __restrict__ W_ih,  // [3D,2D]
                                 const float* __restrict__ W_hh,  // [3D,D]
                                 const float* __restrict__ b_ih,
                                 const float* __restrict__ b_hh,
                                 float* __restrict__ gi,          // [B,3D]
                                 float* __restrict__ gh,          // [B,3D]
                                 int t) {
  const int MT = Bb / 16;        // 4
  const int NT = D3 / 16;        // 96
  int gw = blockIdx.x * (blockDim.x >> 5) + (threadIdx.x >> 5);
  int matsel = gw / (MT * NT);
  int rem    = gw % (MT * NT);
  int mt = rem / NT, nt = rem % NT;

  v8f acc = {};
  if (matsel == 0) {
    // x = concat(emb[:,t,:], ctx); split K=2D into two K=D runs
    const float* A0 = emb + (size_t)(mt * 16) * (Tt * Dh) + (size_t)t * Dh;
    wmma_acc_1x1(acc, A0, Tt * Dh,
                 W_ih + (size_t)(nt * 16) * D2, D2, Dh);
    wmma_acc_1x1(acc, ctx + (size_t)(mt * 16) * Dh, Dh,
                 W_ih + (size_t)(nt * 16) * D2 + Dh, D2, Dh);
    wmma_store_bias(acc, gi + (size_t)(mt * 16) * D3 + nt * 16, D3, b_ih + nt * 16);
  } else {
    wmma_acc_1x1(acc, h + (size_t)(mt * 16) * Dh, Dh,
                 W_hh + (size_t)(nt * 16) * Dh, Dh, Dh);
    wmma_store_bias(acc, gh + (size_t)(mt * 16) * D3 + nt * 16, D3, b_hh + nt * 16);
  }
}

// ---------------------------------------------------------------------------
// Kernel 4 (per step): GRU elementwise update; writes h and outs[b,t,0:D]
// ---------------------------------------------------------------------------
__global__ void gru_update_kernel(const float* __restrict__ gi,
                                  const float* __restrict__ gh,
                                  float* __restrict__ h,
                                  float* __restrict__ outs,   // [B,T,2D]
                                  int t) {
  int i = blockIdx.x * blockDim.x + threadIdx.x;   // over B*D
  int b = i / Dh;
  int d = i - b * Dh;
  const float* gib = gi + (size_t)b * D3;
  const float* ghb = gh + (size_t)b * D3;
  float ir = gib[d],        iz = gib[Dh + d],  in = gib[2 * Dh + d];
  float hr = ghb[d],        hz = ghb[Dh + d],  hn = ghb[2 * Dh + d];
  float r = 1.0f / (1.0f + expf(-(ir + hr)));
  float z = 1.0f / (1.0f + expf(-(iz + hz)));
  float n = tanhf(in + r * hn);
  float hv = h[i];
  float hnew = (1.0f - z) * n + z * hv;
  h[i] = hnew;
  outs[(size_t)b * (Tt * D2) + (size_t)t * D2 + d] = hnew;
}

// ---------------------------------------------------------------------------
// Kernel 5 (per step): q = h @ Wq^T + bq ; 128 single tiles, 1 wave/tile
// ---------------------------------------------------------------------------
__global__ void q_gemm_kernel(const float* __restrict__ h,
                              const float* __restrict__ Wq,
                              const float* __restrict__ bq,
                              float* __restrict__ q) {
  const int NT = Dh / 16;  // 32
  int gw = blockIdx.x * (blockDim.x >> 5) + (threadIdx.x >> 5);
  int mt = gw / NT, nt = gw % NT;
  v8f acc = {};
  wmma_acc_1x1(acc, h + (size_t)(mt * 16) * Dh, Dh,
               Wq + (size_t)(nt * 16) * Dh, Dh, Dh);
  wmma_store_bias(acc, q + (size_t)(mt * 16) * Dh + nt * 16, Dh, bq + nt * 16);
}

// ---------------------------------------------------------------------------
// Kernel 6 (per step): attention — one block per batch row b.
// ---------------------------------------------------------------------------
__global__ void attn_kernel(const float* __restrict__ keys,   // [B,S,D]
                            const float* __restrict__ vals,   // [B,S,D]
                            const unsigned char* __restrict__ mask, // [B,S]
                            const float* __restrict__ q,      // [B,D]
                            float* __restrict__ ctx,          // [B,D]
                            float* __restrict__ outs,         // [B,T,2D]
                            float* __restrict__ attw,         // [B,T,S]
                            int t) {
  __shared__ float qs[Dh];
  __shared__ float sc[Ss];
  __shared__ float red[256];
  int b   = blockIdx.x;
  int tid = threadIdx.x;
  const float scale = 0.044194173824159216f;   // 1/sqrt(512)

  qs[tid]       = q[(size_t)b * Dh + tid];
  qs[tid + 256] = q[(size_t)b * Dh + tid + 256];
  __syncthreads();

  // scores: each thread owns rows s = tid and tid+256; b128 loads
  const float* kb = keys + (size_t)b * Ss * Dh;
  float s0 = 0.0f, s1 = 0.0f;
  {
    const float4* k0 = (const float4*)(kb + (size_t)tid * Dh);
    const float4* k1 = (const float4*)(kb + (size_t)(tid + 256) * Dh);
    const float4* qv = (const float4*)qs;
    for (int d = 0; d < Dh / 4; ++d) {
      float4 qq = qv[d];
      float4 a  = k0[d];
      float4 c  = k1[d];
      s0 += a.x * qq.x + a.y * qq.y + a.z * qq.z + a.w * qq.w;
      s1 += c.x * qq.x + c.y * qq.y + c.z * qq.z + c.w * qq.w;
    }
  }
  s0 *= scale; s1 *= scale;
  const unsigned char* mb = mask + (size_t)b * Ss;
  if (!mb[tid])       s0 = -1e9f;
  if (!mb[tid + 256]) s1 = -1e9f;

  // max-reduce
  red[tid] = fmaxf(s0, s1);
  __syncthreads();
  for (int off = 128; off > 0; off >>= 1) {
    if (tid < off) red[tid] = fmaxf(red[tid], red[tid + off]);
    __syncthreads();
  }
  float mx = red[0];
  __syncthreads();

  float e0 = expf(s0 - mx), e1 = expf(s1 - mx);
  red[tid] = e0 + e1;
  __syncthreads();
  for (int off = 128; off > 0; off >>= 1) {
    if (tid < off) red[tid] += red[tid + off];
    __syncthreads();
  }
  float inv = 1.0f / red[0];
  __syncthreads();

  float w0 = e0 * inv, w1 = e1 * inv;
  sc[tid] = w0; sc[tid + 256] = w1;
  float* aw = attw + (size_t)b * (Tt * Ss) + (size_t)t * Ss;
  aw[tid] = w0; aw[tid + 256] = w1;
  __syncthreads();

  // ctx[d] = sum_s w[s] * values[b,s,d]; lane-coalesced over d
  const float* vb = vals + (size_t)b * Ss * Dh;
  float c0 = 0.0f, c1 = 0.0f;
  for (int s = 0; s < Ss; ++s) {
    float w = sc[s];
    c0 += w * vb[(size_t)s * Dh + tid];
    c1 += w * vb[(size_t)s * Dh + tid + 256];
  }
  ctx[(size_t)b * Dh + tid]       = c0;
  ctx[(size_t)b * Dh + tid + 256] = c1;
  float* ob = outs + (size_t)b * (Tt * D2) + (size_t)t * D2 + Dh;
  ob[tid] = c0; ob[tid + 256] = c1;
}

// ---------------------------------------------------------------------------
// Kernel 7: logits = outs @ Wo^T + bo
//   [B*T=16384, 2D] x [V=4000, 2D]; wave computes 64x32 (4 M x 2 N subtiles)
// ---------------------------------------------------------------------------
__global__ void logits_gemm_kernel(const float* __restrict__ outs,  // [B*T, 2D]
                                   const float* __restrict__ Wo,    // [V, 2D]
                                   const float* __restrict__ bo,
                                   float* __restrict__ logits) {    // [B*T, V]
  const int MG = (Bb * Tt) / 64;  // 256 M-groups (64 rows each)
  const int NG = Vv / 32;         // 125 N-groups (32 cols each)
  int gw = blockIdx.x * (blockDim.x >> 5) + (threadIdx.x >> 5);
  int mg = gw / NG, ng = gw % NG;

  v8f acc[4][2] = {{{}, {}}, {{}, {}}, {{}, {}}, {{}, {}}};
  wmma_acc_4x2(acc, outs + (size_t)(mg * 64) * D2, D2,
               Wo + (size_t)(ng * 32) * D2, D2, D2);
#pragma unroll
  for (int mj = 0; mj < 4; ++mj) {
#pragma unroll
    for (int nj = 0; nj < 2; ++nj) {
      int row0 = mg * 64 + mj * 16;
      int col0 = ng * 32 + nj * 16;
      wmma_store_bias(acc[mj][nj], logits + (size_t)row0 * Vv + col0, Vv, bo + col0);
    }
  }
}

// ---------------------------------------------------------------------------
// Host launcher
// ---------------------------------------------------------------------------
extern "C" void kernel_launch(void* const* d_in, const int* in_sizes, int n_in,
                              void* d_out, int out_size, void* d_ws, size_t ws_size,
                              hipStream_t stream) {
  const int*           input    = (const int*)d_in[0];
  const float*         features = (const float*)d_in[1];
  const unsigned char* fmask    = (const unsigned char*)d_in[2];
  const float*         emb_W    = (const float*)d_in[3];
  const float*         W_ih     = (const float*)d_in[4];
  const float*         W_hh     = (const float*)d_in[5];
  const float*         b_ih     = (const float*)d_in[6];
  const float*         b_hh     = (const float*)d_in[7];
  const float*         Wq       = (const float*)d_in[8];
  const float*         bq       = (const float*)d_in[9];
  const float*         Wk       = (const float*)d_in[10];
  const float*         bk       = (const float*)d_in[11];
  const float*         Wv       = (const float*)d_in[12];
  const float*         bv       = (const float*)d_in[13];
  const float*         Wo       = (const float*)d_in[14];
  const float*         bo       = (const float*)d_in[15];

  float* out    = (float*)d_out;
  float* logits = out;                                     // [B,T,V]
  float* attw   = out + (size_t)Bb * Tt * Vv;              // [B,1,T,S]

  // workspace carve-up (floats)
  float* ws   = (float*)d_ws;
  float* keys = ws;                                        size_t off = (size_t)Bb * Ss * Dh;
  float* vals = ws + off;                                  off += (size_t)Bb * Ss * Dh;
  float* emb  = ws + off;                                  off += (size_t)Bb * Tt * Dh;
  float* outs = ws + off;                                  off += (size_t)Bb * Tt * D2;
  float* h    = ws + off;                                  off += (size_t)Bb * Dh;
  float* ctx  = ws + off;                                  off += (size_t)Bb * Dh;
  float* q    = ws + off;                                  off += (size_t)Bb * Dh;
  float* gi   = ws + off;                                  off += (size_t)Bb * D3;
  float* gh   = ws + off;                                  off += (size_t)Bb * D3;

  // h and ctx are contiguous -> one zero pass (carry state for t=0)
  init_zero_kernel<<<(2 * Bb * Dh + 255) / 256, 256, 0, stream>>>(h, 2 * Bb * Dh);

  // K/V projections: 2 mats * 2048 M-tiles * 8 N-groups = 32768 waves
  kv_proj_kernel<<<(2 * ((Bb * Ss) / 16) * (Dh / 64)) / 8, 256, 0, stream>>>(
      features, Wk, bk, Wv, bv, keys, vals);

  // embedding gather
  gather_emb_kernel<<<(Bb * Tt * Dh) / 256, 256, 0, stream>>>(input, emb_W, emb);

  // sequential recurrence over T
  for (int t = 0; t < Tt; ++t) {
    gate_gemm_kernel<<<(2 * (Bb / 16) * (D3 / 16)) / 8, 256, 0, stream>>>(
        emb, ctx, h, W_ih, W_hh, b_ih, b_hh, gi, gh, t);
    gru_update_kernel<<<(Bb * Dh) / 256, 256, 0, stream>>>(gi, gh, h, outs, t);
    q_gemm_kernel<<<((Bb / 16) * (Dh / 16)) / 8, 256, 0, stream>>>(h, Wq, bq, q);
    attn_kernel<<<Bb, 256, 0, stream>>>(keys, vals, fmask, q, ctx, outs, attw, t);
  }

  // output projection: 256 M-groups * 125 N-groups = 32000 waves
  logits_gemm_kernel<<<(((Bb * Tt) / 64) * (Vv / 32)) / 8, 256, 0, stream>>>(
      outs, Wo, bo, logits);
}